// LoRALayer_17102559772997
// MI455X (gfx1250) — compile-verified
//
#include <hip/hip_runtime.h>
#include <stdint.h>

// ---------------------------------------------------------------------------
// LoRA fused forward for MI455X (gfx1250, wave32).
//   t = x @ A^T   (M=16384, K=4096, N=16),  y = dropout(scaling * t @ B^T)
// Memory-bound: 268MB read + 268MB write @ 23.3 TB/s ~ 23us floor.
// One workgroup per 16-token tile; stream x once; double-buffered LDS staging
// via GLOBAL_LOAD_ASYNC_TO_LDS_B128 (ASYNCcnt-tracked DMA); WMMA f32 16x16x4
// for both GEMMs; rank-16 intermediate kept in LDS.
// ---------------------------------------------------------------------------

typedef float v2f __attribute__((ext_vector_type(2)));
typedef float v4f __attribute__((ext_vector_type(4)));
typedef float v8f __attribute__((ext_vector_type(8)));

#define IN_F     4096
#define OUT_F    4096
#define RANK     16
#define SCALING  2.0f            // alpha/rank = 32/16
#define KEEP     0.9f
#define INV_KEEP (1.0f / 0.9f)

#define KC   128                 // K chunk staged in LDS
#define NC   128                 // output-column chunk
#define NKC  (IN_F / KC)         // 32
#define NOC  (OUT_F / NC)        // 32

// LDS strides: multiples of 4 floats -> every float4 slot is 16B aligned
// (required for b128 LDS writes) and rows spread across the 64 banks.
#define XS_LD 132                // bank stride 132%64 = 4 -> 16 distinct banks
#define BS_LD 20                 // bank stride 20 -> 16 distinct banks
#define TT_LD 18

#if __has_builtin(__builtin_amdgcn_global_load_async_to_lds_b128)
#define HAVE_ASYNC_LDS 1
#else
#define HAVE_ASYNC_LDS 0
#endif

#if HAVE_ASYNC_LDS
// Builtin prototype (from diagnostic): (v4i AS1* src, v4i AS3* dst, Ii, Ii)
typedef int v4i __attribute__((__vector_size__(16)));
typedef __attribute__((address_space(1))) v4i* gv4i_p;
typedef __attribute__((address_space(3))) v4i* lv4i_p;
__device__ __forceinline__ void wait_async0() {
#if __has_builtin(__builtin_amdgcn_s_wait_asynccnt)
    __builtin_amdgcn_s_wait_asynccnt(0);
#else
    asm volatile("s_wait_asynccnt 0" ::: "memory");
#endif
}
#else
__device__ __forceinline__ void wait_async0() {}
#endif

// Copy one float4 global -> LDS (async DMA when available).
__device__ __forceinline__ void cp16(const float* __restrict__ g, float* l) {
#if HAVE_ASYNC_LDS
    // Generic->AS casts via uintptr_t: global value unchanged; LDS generic
    // address' low 32 bits are the LDS offset (ISA 10.2 aperture rules).
    __builtin_amdgcn_global_load_async_to_lds_b128(
        (gv4i_p)(uintptr_t)g, (lv4i_p)(uintptr_t)l, 0, 0);
#else
    *(v4f*)l = *(const v4f*)g;        // global_load_b128 + ds_store_b128
#endif
}

// Stage x-tile chunk [16][KC] and A chunk [16][KC] (coalesced float4 per thread).
__device__ __forceinline__ void stage_xa(const float* __restrict__ x,
                                         const float* __restrict__ Am,
                                         float* xsb, float* asb,
                                         int tid, int m0, int k0) {
    #pragma unroll
    for (int j = 0; j < 2; ++j) {
        const int q   = tid + j * 256;     // float4 slot 0..511
        const int r   = q >> 5;            // row 0..15 (32 float4 per row)
        const int k4i = (q & 31) << 2;     // 0..124
        cp16(x  + (size_t)(m0 + r) * IN_F + k0 + k4i, &xsb[r * XS_LD + k4i]);
        cp16(Am + (size_t)r        * IN_F + k0 + k4i, &asb[r * XS_LD + k4i]);
#if !HAVE_ASYNC_LDS
        __builtin_prefetch(x + (size_t)(m0 + r) * IN_F + k0 + KC + k4i, 0, 1);
#endif
    }
}

// Stage B chunk: contiguous slab of lora_B [NC][RANK] -> fully coalesced.
__device__ __forceinline__ void stage_b(const float* __restrict__ Bm,
                                        float* bsb, int tid, int o0) {
    #pragma unroll
    for (int j = 0; j < 2; ++j) {
        const int q  = tid + j * 256;      // float4 slot, 4 per out row
        const int o  = q >> 2;             // 0..127
        const int r4 = (q & 3) << 2;       // 0,4,8,12
        cp16(Bm + (size_t)(o0 + o) * RANK + r4, &bsb[o * BS_LD + r4]);
    }
}

__device__ __forceinline__ float hash_u01(uint32_t idx) {
    uint32_t h = idx * 0x9E3779B1u + 42u;
    h ^= h >> 16;  h *= 0x7FEB352Du;
    h ^= h >> 15;  h *= 0x846CA68Bu;
    h ^= h >> 16;
    return (float)(h >> 8) * (1.0f / 16777216.0f);
}

// Phase-exclusive LDS regions (separated by barriers) share one union.
struct PhaseA { float xs[2][16 * XS_LD]; float as[2][16 * XS_LD]; }; // 33 KB
struct PhaseR { float tts[8 * 256]; };                               //  8 KB
struct PhaseB { float bs[2][NC * BS_LD]; };                          // 20 KB

__global__ __launch_bounds__(256)
void lora_fused_kernel(const float* __restrict__ x,
                       const float* __restrict__ Am,   // [RANK][IN_F]
                       const float* __restrict__ Bm,   // [OUT_F][RANK]
                       float* __restrict__ out) {
    __shared__ union { PhaseA a; PhaseR r; PhaseB b; } sm;
    __shared__ float tt[16 * TT_LD];       // final t tile [16 tokens][16 ranks]

    const int tid   = threadIdx.x;
    const int wave  = tid >> 5;
    const int lane  = tid & 31;
    const int m0    = blockIdx.x * 16;

    // WMMA f32 16x16x4 operand lane mapping (ISA 7.12.2):
    //   lanes 0-15 -> M/N = 0..15 with K{0,1} in v0,v1; lanes 16-31 -> K{2,3}.
    const int row16 = lane & 15;
    const int hi    = lane >> 4;
    const int kk    = hi * 2;

    // ---------------- Phase A: t = x @ A^T (double-buffered) ----------------
    v8f acc = {0.f, 0.f, 0.f, 0.f, 0.f, 0.f, 0.f, 0.f};

    stage_xa(x, Am, sm.a.xs[0], sm.a.as[0], tid, m0, 0);          // prologue

    for (int c = 0; c < NKC; ++c) {
        const int cur = c & 1;
        wait_async0();                     // own chunk-c copies committed
        __syncthreads();                   // everyone's chunk-c data visible;
                                           // buf cur^1 fully drained of readers
        if (c + 1 < NKC)
            stage_xa(x, Am, sm.a.xs[cur ^ 1], sm.a.as[cur ^ 1],
                     tid, m0, (c + 1) * KC);   // streams during compute below

        const float* xsb = sm.a.xs[cur];
        const float* asb = sm.a.as[cur];
        const int kbase = wave * 16;       // 8 waves split KC=128
        #pragma unroll
        for (int s = 0; s < 4; ++s) {
            const int k4 = kbase + s * 4;
            v2f aop = *(const v2f*)&xsb[row16 * XS_LD + k4 + kk]; // x[token][k]
            v2f bop = *(const v2f*)&asb[row16 * XS_LD + k4 + kk]; // (A^T)[k][rank]
            acc = __builtin_amdgcn_wmma_f32_16x16x4_f32(
                      false, aop, false, bop, (short)0, acc, false, false);
        }
    }

    // Cross-wave reduction of partial 16x16 t tiles (aliases phase-A LDS).
    __syncthreads();                       // all reads of sm.a done
    #pragma unroll
    for (int r = 0; r < 8; ++r)
        sm.r.tts[wave * 256 + (r + 8 * hi) * 16 + row16] = acc[r];
    __syncthreads();
    {
        float ssum = 0.f;
        #pragma unroll
        for (int w = 0; w < 8; ++w) ssum += sm.r.tts[w * 256 + tid];
        tt[(tid >> 4) * TT_LD + (tid & 15)] = ssum * SCALING;
    }
    __syncthreads();                       // tt ready; tts region free for bs

    // ---------------- Phase B: y = dropout(t @ B^T) (double-buffered) -------
    stage_b(Bm, sm.b.bs[0], tid, 0);       // prologue
    const int wbase = wave * 16;

    for (int oc = 0; oc < NOC; ++oc) {
        const int cur = oc & 1;
        wait_async0();
        __syncthreads();
        if (oc + 1 < NOC)
            stage_b(Bm, sm.b.bs[cur ^ 1], tid, (oc + 1) * NC);

        const float* bsb = sm.b.bs[cur];
        v8f acc2 = {0.f, 0.f, 0.f, 0.f, 0.f, 0.f, 0.f, 0.f};
        #pragma unroll
        for (int s = 0; s < 4; ++s) {      // K = RANK = 16 -> 4 steps
            const int k4 = s * 4;
            v2f aop = *(const v2f*)&tt[row16 * TT_LD + k4 + kk];           // t[token][rank]
            v2f bop = *(const v2f*)&bsb[(wbase + row16) * BS_LD + k4 + kk];// (B^T)[rank][out]
            acc2 = __builtin_amdgcn_wmma_f32_16x16x4_f32(
                       false, aop, false, bop, (short)0, acc2, false, false);
        }

        // Inverted dropout + store (C/D layout: VGPR r -> token M=r+8*hi).
        const int ocol = oc * NC + wbase + row16;
        #pragma unroll
        for (int r = 0; r < 8; ++r) {
            const int tok = r + 8 * hi;
            const size_t oidx = (size_t)(m0 + tok) * OUT_F + ocol;
            const float u = hash_u01((uint32_t)oidx);
            out[oidx] = (u < KEEP) ? acc2[r] * INV_KEEP : 0.0f;
        }
    }
}

extern "C" void kernel_launch(void* const* d_in, const int* in_sizes, int n_in,
                              void* d_out, int out_size, void* d_ws, size_t ws_size,
                              hipStream_t stream) {
    (void)n_in; (void)out_size; (void)d_ws; (void)ws_size;
    const float* x  = (const float*)d_in[0];   // [4,4096,4096] f32
    const float* Am = (const float*)d_in[1];   // [16,4096]     f32
    const float* Bm = (const float*)d_in[2];   // [4096,16]     f32
    float* out = (float*)d_out;                // [4,4096,4096] f32

    const int rows   = in_sizes[0] / IN_F;     // 16384 token rows
    const int blocks = rows / 16;              // one 16-token tile per workgroup

    hipLaunchKernelGGL(lora_fused_kernel, dim3(blocks), dim3(256), 0, stream,
                       x, Am, Bm, out);
}